// EdgeGuidedModule_65343632441930
// MI455X (gfx1250) — compile-verified
//
#include <hip/hip_runtime.h>
#include <stdint.h>
#include <math.h>

// Problem constants (from reference: image [4,1,128,256,256] f32)
namespace {
constexpr int Dd = 128, Hh = 256, Ww = 256, Bb = 4;
constexpr int TILE = 32;           // H x W tile per block
constexpr int CD   = 16;           // D-slices per block
constexpr int ROWS = TILE + 2;     // 34 halo'd rows per plane
constexpr int LSTR = 40;           // LDS row stride in floats (10 x float4, 16B-aligned vectors)
constexpr int NVEC = ROWS * 10;    // 340 float4 vectors per plane
constexpr int PLANE = ROWS * LSTR; // 1360 floats per LDS plane buffer
}

// CDNA5 async global->LDS copies (GVS mode: 64-bit SGPR base + 32-bit VGPR offset).
// Tracked by ASYNCcnt; data never touches VGPRs.
__device__ __forceinline__ void async_load_b128(uint32_t lds_byte_addr,
                                                int32_t voff_bytes,
                                                uint64_t sbase) {
  asm volatile("global_load_async_to_lds_b128 %0, %1, %2"
               :
               : "v"(lds_byte_addr), "v"(voff_bytes), "s"(sbase)
               : "memory");
}

__device__ __forceinline__ void wait_async() {
  asm volatile("s_wait_asynccnt 0" ::: "memory");
}

__global__ __launch_bounds__(256)
void sobel3d_mag_kernel(const float* __restrict__ img, float* __restrict__ out) {
  __shared__ alignas(16) float sm[2][PLANE];

  const int tid = threadIdx.x;
  const int tx  = (blockIdx.x & 7) * TILE;    // W tile origin (multiple of 32)
  const int ty  = (blockIdx.x >> 3) * TILE;   // H tile origin
  const int z0  = blockIdx.y * CD;            // D chunk origin
  const int b   = blockIdx.z;

  const float* inb = img + (size_t)b * Dd * Hh * Ww;

  // ---- plane fetch: async b128 copy of one halo'd plane into an LDS buffer ----
  // LDS col 0 maps to gx = tx-4; row r maps to gy = ty+r-1. Every 16B vector is
  // either fully inside the image row or fully outside (zero-filled = conv pad).
  auto plane_load = [&](int z, int buf) {
    const bool zin = (z >= 0) && (z < Dd);
    const uint64_t sbase =
        zin ? (uint64_t)(uintptr_t)(inb + (size_t)z * Hh * Ww) : 0ull;
#pragma unroll
    for (int k = 0; k < 2; ++k) {
      int idx = tid + k * 256;
      if (idx < NVEC) {
        int r = idx / 10, v = idx - r * 10;
        int gy  = ty + r - 1;
        int gxv = tx - 4 + v * 4;            // 16B-aligned vector start
        float* dst = &sm[buf][r * LSTR + v * 4];
        if (zin && (unsigned)gy < (unsigned)Hh && (unsigned)gxv < (unsigned)Ww) {
          async_load_b128((uint32_t)(uintptr_t)dst, (gy * Ww + gxv) * 4, sbase);
        } else {
          *(float4*)dst = make_float4(0.f, 0.f, 0.f, 0.f);  // zero padding
        }
      }
    }
  };

  // ---- per-pixel rolling registers over 3 D-planes ----
  const int x  = tid & 31;   // tile-local x (wave lanes -> consecutive x)
  const int yb = tid >> 5;   // 0..7; thread owns rows yb, yb+8, yb+16, yb+24

  float Ap[4], Ac[4], An[4];   // smoothH(smoothW) -> Gz via D-derivative
  float Bp[4], Bc[4], Bn[4];   // smoothH(derivW)  -> Gx via D-smooth
  float Cp[4], Cc[4], Cn[4];   // derivH(smoothW)  -> Gy via D-smooth
  float Ec[4], En[4];          // center value (channel-0 copy)

  // Separable in-plane pass: 9 LDS reads per pixel.
  auto plane_calc = [&](int buf) {
#pragma unroll
    for (int i = 0; i < 4; ++i) {
      int ly = yb + 8 * i + 1;    // center LDS row
      int lx = x + 4;             // center LDS col
      const float* r0 = &sm[buf][(ly - 1) * LSTR + lx];
      const float* r1 = &sm[buf][(ly    ) * LSTR + lx];
      const float* r2 = &sm[buf][(ly + 1) * LSTR + lx];
      float a0 = r0[-1], a1 = r0[0], a2 = r0[1];
      float b0 = r1[-1], b1 = r1[0], b2 = r1[1];
      float c0 = r2[-1], c1 = r2[0], c2 = r2[1];
      float s0 = a0 + 2.0f * a1 + a2;   // smooth-W per H row
      float s1 = b0 + 2.0f * b1 + b2;
      float s2 = c0 + 2.0f * c1 + c2;
      float d0 = a2 - a0;               // deriv-W per H row
      float d1 = b2 - b0;
      float d2 = c2 - c0;
      An[i] = s0 + 2.0f * s1 + s2;
      Cn[i] = s2 - s0;
      Bn[i] = d0 + 2.0f * d1 + d2;
      En[i] = b1;
    }
  };

  // ---- software pipeline prologue ----
  plane_load(z0 - 1, 0);
  wait_async();
  __syncthreads();

  plane_load(z0, 1);            // overlap: fetch z0 while computing z0-1
  plane_calc(0);
#pragma unroll
  for (int i = 0; i < 4; ++i) { Ap[i] = An[i]; Bp[i] = Bn[i]; Cp[i] = Cn[i]; }
  wait_async();
  __syncthreads();

  plane_load(z0 + 1, 0);        // overlap: fetch z0+1 while computing z0
  plane_calc(1);
#pragma unroll
  for (int i = 0; i < 4; ++i) { Ac[i] = An[i]; Bc[i] = Bn[i]; Cc[i] = Cn[i]; Ec[i] = En[i]; }
  wait_async();
  __syncthreads();

  float* outb = out + (size_t)b * 2u * Dd * Hh * Ww;
  const size_t chstride = (size_t)Dd * Hh * Ww;

  // ---- main loop over output slices ----
  for (int z = z0; z < z0 + CD; ++z) {
    int nb = (z - z0) & 1;                            // buffer holding plane z+1
    if (z + 2 <= z0 + CD) plane_load(z + 2, nb ^ 1);  // prefetch two ahead
    plane_calc(nb);                                   // intermediates for plane z+1

#pragma unroll
    for (int i = 0; i < 4; ++i) {
      float Gx = Bp[i] + 2.0f * Bc[i] + Bn[i];
      float Gy = Cp[i] + 2.0f * Cc[i] + Cn[i];
      float Gz = An[i] - Ap[i];
      float mag = sqrtf(Gx * Gx + Gy * Gy + Gz * Gz + 1e-8f);

      size_t o = ((size_t)z * Hh + (size_t)(ty + yb + 8 * i)) * Ww + (size_t)(tx + x);
      __builtin_nontemporal_store(Ec[i], outb + o);             // channel 0: copy
      __builtin_nontemporal_store(mag,   outb + chstride + o);  // channel 1: |G|

      Ap[i] = Ac[i]; Ac[i] = An[i];
      Bp[i] = Bc[i]; Bc[i] = Bn[i];
      Cp[i] = Cc[i]; Cc[i] = Cn[i];
      Ec[i] = En[i];
    }
    wait_async();
    __syncthreads();
  }
}

extern "C" void kernel_launch(void* const* d_in, const int* in_sizes, int n_in,
                              void* d_out, int out_size, void* d_ws, size_t ws_size,
                              hipStream_t stream) {
  (void)in_sizes; (void)n_in; (void)d_ws; (void)ws_size; (void)out_size;
  const float* img = (const float*)d_in[0];
  float* out = (float*)d_out;

  dim3 grid((Hh / TILE) * (Ww / TILE),  // 64 HxW tiles
            Dd / CD,                    // 8 D-chunks
            Bb);                        // 4 batches
  dim3 block(256);
  sobel3d_mag_kernel<<<grid, block, 0, stream>>>(img, out);
}